// COSNCELoss_72009421684964
// MI455X (gfx1250) — compile-verified
//
#include <hip/hip_runtime.h>
#include <hip/hip_bf16.h>
#include <math.h>

// Problem constants (from reference): b=64, seq=2048, hid=256, f32 in/out.
#define B_    64
#define SEQ_  2048
#define HID_  256
#define ROWSTRIDE_ ((size_t)SEQ_ * (size_t)HID_)   // stride between batch rows (floats)

typedef float v2f __attribute__((ext_vector_type(2)));
typedef float v8f __attribute__((ext_vector_type(8)));

#if __has_builtin(__builtin_amdgcn_global_load_async_to_lds_b128) && \
    __has_builtin(__builtin_amdgcn_s_wait_asynccnt)
#define HAS_ASYNC_LDS 1
typedef int v4i __attribute__((ext_vector_type(4)));
typedef __attribute__((address_space(1))) v4i* g4p;   // global src for async load
typedef __attribute__((address_space(3))) v4i* l4p;   // LDS dst for async load
#endif

__device__ __forceinline__ float dot4f(const float4& a, const float4& b) {
  return fmaf(a.x, b.x, fmaf(a.y, b.y, fmaf(a.z, b.z, a.w * b.w)));
}

// ---------------------------------------------------------------------------
// Phase 1: one block per seq position s. 8 waves; each wave handles 8 of the
// 64 batch rows. Rows are staged global->LDS with double-buffered
// global_load_async_to_lds_b128 (ASYNCcnt pipelined), then each lane computes
// on its 8 floats. Produces per-s partials:
//   ws[s]        = sum_i <rsn[i,s], rtn[i,s]>          (diagonal)
//   ws[SEQ + s]  = <sum_i rsn[i,s], sum_j rtn[j,s]>    (all-pairs)
// Deterministic: shuffle + LDS tree reductions only, no atomics.
// ---------------------------------------------------------------------------
__global__ __launch_bounds__(256) void cosnce_phase1(
    const float* __restrict__ r_s, const float* __restrict__ r_t,
    float* __restrict__ ws) {
  const int s    = blockIdx.x;
  const int wave = threadIdx.x >> 5;
  const int lane = threadIdx.x & 31;

  __shared__ float sh_u[8][HID_];
  __shared__ float sh_v[8][HID_];
  __shared__ float sh_d[8];
  __shared__ float sh_p[8];
#ifdef HAS_ASYNC_LDS
  __shared__ float stage[8][2][2][HID_];   // [wave][buffer][tensor][256 floats] = 32 KB
#endif

  float ua[8], va[8];
#pragma unroll
  for (int k = 0; k < 8; ++k) { ua[k] = 0.f; va[k] = 0.f; }
  float diag = 0.f;

  const size_t rowbase = (size_t)s * HID_;

#ifdef HAS_ASYNC_LDS
  // ---- async double-buffered staging path (gfx1250 ASYNCcnt pipeline) ----
  {
    // Preload ii=0 into buffer 0: 2 x b128 per row (16B/lane x 32 lanes = 512B).
    const float* ps = r_s + (size_t)wave * ROWSTRIDE_ + rowbase;
    const float* pt = r_t + (size_t)wave * ROWSTRIDE_ + rowbase;
    float* ds = &stage[wave][0][0][0];
    float* dt = &stage[wave][0][1][0];
    __builtin_amdgcn_global_load_async_to_lds_b128((g4p)(ps +       lane * 4), (l4p)(ds +       lane * 4), 0, 0);
    __builtin_amdgcn_global_load_async_to_lds_b128((g4p)(ps + 128 + lane * 4), (l4p)(ds + 128 + lane * 4), 0, 0);
    __builtin_amdgcn_global_load_async_to_lds_b128((g4p)(pt +       lane * 4), (l4p)(dt +       lane * 4), 0, 0);
    __builtin_amdgcn_global_load_async_to_lds_b128((g4p)(pt + 128 + lane * 4), (l4p)(dt + 128 + lane * 4), 0, 0);
  }
  for (int ii = 0; ii < 8; ++ii) {
    const int buf = ii & 1;
    if (ii < 7) {
      // Issue next row-pair into the other buffer, then wait for current one.
      const int inext = ((ii + 1) << 3) + wave;
      const float* ps = r_s + (size_t)inext * ROWSTRIDE_ + rowbase;
      const float* pt = r_t + (size_t)inext * ROWSTRIDE_ + rowbase;
      float* ds = &stage[wave][buf ^ 1][0][0];
      float* dt = &stage[wave][buf ^ 1][1][0];
      __builtin_amdgcn_global_load_async_to_lds_b128((g4p)(ps +       lane * 4), (l4p)(ds +       lane * 4), 0, 0);
      __builtin_amdgcn_global_load_async_to_lds_b128((g4p)(ps + 128 + lane * 4), (l4p)(ds + 128 + lane * 4), 0, 0);
      __builtin_amdgcn_global_load_async_to_lds_b128((g4p)(pt +       lane * 4), (l4p)(dt +       lane * 4), 0, 0);
      __builtin_amdgcn_global_load_async_to_lds_b128((g4p)(pt + 128 + lane * 4), (l4p)(dt + 128 + lane * 4), 0, 0);
      __builtin_amdgcn_s_wait_asynccnt(4);   // in-order: oldest 4 (current buf) done
    } else {
      __builtin_amdgcn_s_wait_asynccnt(0);
    }
    const float* cs = &stage[wave][buf][0][lane << 3];
    const float* ct = &stage[wave][buf][1][lane << 3];
    const float4 a0 = ((const float4*)cs)[0];
    const float4 a1 = ((const float4*)cs)[1];
    const float4 b0 = ((const float4*)ct)[0];
    const float4 b1 = ((const float4*)ct)[1];
#else
  // ---- fallback: direct b128 loads to VGPRs ----
  for (int ii = 0; ii < 8; ++ii) {
    const int i = (ii << 3) + wave;
    const float* ps = r_s + (size_t)i * ROWSTRIDE_ + rowbase + (size_t)(lane << 3);
    const float* pt = r_t + (size_t)i * ROWSTRIDE_ + rowbase + (size_t)(lane << 3);
    if (ii < 7) {
      __builtin_prefetch(ps + 8 * ROWSTRIDE_, 0, 0);
      __builtin_prefetch(pt + 8 * ROWSTRIDE_, 0, 0);
    }
    const float4 a0 = ((const float4*)ps)[0];
    const float4 a1 = ((const float4*)ps)[1];
    const float4 b0 = ((const float4*)pt)[0];
    const float4 b1 = ((const float4*)pt)[1];
#endif

    float ss = dot4f(a0, a0) + dot4f(a1, a1);
    float st = dot4f(b0, b0) + dot4f(b1, b1);
    float sd = dot4f(a0, b0) + dot4f(a1, b1);
#pragma unroll
    for (int m = 16; m >= 1; m >>= 1) {      // wave32 butterfly reduce
      ss += __shfl_xor(ss, m, 32);
      st += __shfl_xor(st, m, 32);
      sd += __shfl_xor(sd, m, 32);
    }
    const float inv_s = 1.0f / fmaxf(sqrtf(ss), 1e-12f);
    const float inv_t = 1.0f / fmaxf(sqrtf(st), 1e-12f);
    diag = fmaf(sd, inv_s * inv_t, diag);

    ua[0] = fmaf(a0.x, inv_s, ua[0]);  ua[1] = fmaf(a0.y, inv_s, ua[1]);
    ua[2] = fmaf(a0.z, inv_s, ua[2]);  ua[3] = fmaf(a0.w, inv_s, ua[3]);
    ua[4] = fmaf(a1.x, inv_s, ua[4]);  ua[5] = fmaf(a1.y, inv_s, ua[5]);
    ua[6] = fmaf(a1.z, inv_s, ua[6]);  ua[7] = fmaf(a1.w, inv_s, ua[7]);
    va[0] = fmaf(b0.x, inv_t, va[0]);  va[1] = fmaf(b0.y, inv_t, va[1]);
    va[2] = fmaf(b0.z, inv_t, va[2]);  va[3] = fmaf(b0.w, inv_t, va[3]);
    va[4] = fmaf(b1.x, inv_t, va[4]);  va[5] = fmaf(b1.y, inv_t, va[5]);
    va[6] = fmaf(b1.z, inv_t, va[6]);  va[7] = fmaf(b1.w, inv_t, va[7]);
  }

#pragma unroll
  for (int k = 0; k < 8; ++k) {
    sh_u[wave][(lane << 3) + k] = ua[k];
    sh_v[wave][(lane << 3) + k] = va[k];
  }
  if (lane == 0) sh_d[wave] = diag;
  __syncthreads();

  // Combine the 8 per-wave partial sums; each thread owns one hid channel.
  const int h = threadIdx.x;
  float u = 0.f, v = 0.f;
#pragma unroll
  for (int w = 0; w < 8; ++w) { u += sh_u[w][h]; v += sh_v[w][h]; }
  float p = u * v;
#pragma unroll
  for (int m = 16; m >= 1; m >>= 1) p += __shfl_xor(p, m, 32);
  if (lane == 0) sh_p[wave] = p;
  __syncthreads();

  if (threadIdx.x == 0) {
    float s_all = 0.f, s_diag = 0.f;
#pragma unroll
    for (int w = 0; w < 8; ++w) { s_all += sh_p[w]; s_diag += sh_d[w]; }
    ws[s]        = s_diag;
    ws[SEQ_ + s] = s_all;
  }
}

// ---------------------------------------------------------------------------
// Phase 2: one wave reduces the 2x2048 per-s partials with V_WMMA_F32_16X16X4.
// Row-sum trick: D = A x Ones, A = 64 partials in the ISA 16x4 layout
// (lane m holds A[m, 0..1], lane m+16 holds A[m, 2..3]); every column of D
// then holds the 16 row sums; accumulate over C, finish with shfl_xor(16).
// Exact in f32. Then apply the closed-form loss.
// ---------------------------------------------------------------------------
__global__ __launch_bounds__(32) void cosnce_phase2(
    const float* __restrict__ ws, float* __restrict__ out) {
  const int lane = threadIdx.x & 31;
  const int m  = lane & 15;
  const int kb = (lane >> 4) << 1;

  v2f ones; ones[0] = 1.0f; ones[1] = 1.0f;
  v8f acc_d = {};
  v8f acc_a = {};

  for (int t = 0; t < SEQ_; t += 64) {
    const int idx = t + (m << 2) + kb;
    v2f ad; ad[0] = ws[idx];         ad[1] = ws[idx + 1];
    v2f aa; aa[0] = ws[SEQ_ + idx];  aa[1] = ws[SEQ_ + idx + 1];
    acc_d = __builtin_amdgcn_wmma_f32_16x16x4_f32(
        false, ad, false, ones, (short)0, acc_d, false, false);
    acc_a = __builtin_amdgcn_wmma_f32_16x16x4_f32(
        false, aa, false, ones, (short)0, acc_a, false, false);
  }

  float cd = 0.f, ca = 0.f;
#pragma unroll
  for (int r = 0; r < 8; ++r) { cd += acc_d[r]; ca += acc_a[r]; }
  cd += __shfl_xor(cd, 16, 32);   // rows 0-7 + rows 8-15 of column `lane`
  ca += __shfl_xor(ca, 16, 32);

  if (lane == 0) {
    const float K = (float)(B_ - 1);
    const float loss_sum = 2.0f * (float)(B_ * SEQ_)
                         - (1.5f + 0.5f / K) * cd
                         + (0.5f / K) * ca;
    out[0] = loss_sum / ((float)B_ * 200.0f * (K + 1.0f) * 4.0f);
  }
}

extern "C" void kernel_launch(void* const* d_in, const int* in_sizes, int n_in,
                              void* d_out, int out_size, void* d_ws, size_t ws_size,
                              hipStream_t stream) {
  const float* r_s = (const float*)d_in[0];
  const float* r_t = (const float*)d_in[1];
  float* ws  = (float*)d_ws;      // 2*SEQ_ floats = 16 KB of scratch
  float* out = (float*)d_out;     // 1 float

  cosnce_phase1<<<SEQ_, 256, 0, stream>>>(r_s, r_t, ws);
  cosnce_phase2<<<1, 32, 0, stream>>>(ws, out);
}